// GAEEncoder_7919919694018
// MI455X (gfx1250) — compile-verified
//
#include <hip/hip_runtime.h>
#include <stdint.h>
#include <stddef.h>

// GAE encoder (3x GCNConv) for MI455X / gfx1250, wave32.
// GEMMs use V_WMMA_F32_16X16X4_F32 (native f32 -> matches reference precision).
// Edge aggregation = segment_sum via global_atomic_add_f32 (L2-resident).

typedef float v2f __attribute__((ext_vector_type(2)));
typedef float v8f __attribute__((ext_vector_type(8)));

#define HD 64   // hidden dim

__device__ __forceinline__ void atomic_add_f32(float* p, float v) {
    uint64_t a = (uint64_t)(uintptr_t)p;
    asm volatile("global_atomic_add_f32 %0, %1, off" : : "v"(a), "v"(v) : "memory");
}

// ---------------- degree / normalization ----------------

__global__ __launch_bounds__(256) void k_init_deg(float* deg, int n) {
    int i = blockIdx.x * 256 + threadIdx.x;
    if (i < n) deg[i] = 1.0f;               // self-loop contribution
}

__global__ __launch_bounds__(256) void k_deg(const long long* __restrict__ dst,
                                             float* deg, int e) {
    int i = blockIdx.x * 256 + threadIdx.x;
    if (i < e) atomic_add_f32(&deg[(int)dst[i]], 1.0f);
}

__global__ __launch_bounds__(256) void k_dis(const float* __restrict__ deg,
                                             float* dis, int n) {
    int i = blockIdx.x * 256 + threadIdx.x;
    if (i < n) dis[i] = rsqrtf(deg[i]);     // deg >= 1 always (self-loop)
}

__global__ __launch_bounds__(256) void k_zero(float* p, size_t n) {
    size_t i = (size_t)blockIdx.x * 256 + threadIdx.x;
    size_t stride = (size_t)gridDim.x * 256;
    for (; i < n; i += stride) p[i] = 0.0f;
}

// ---------------- dense GEMM: out[N,64] = in[N,K] @ W[K,64] ----------------
// 4 waves / block; each wave -> 16 rows x 64 cols (four 16x16 f32 tiles).
// V_WMMA_F32_16X16X4_F32 operand layouts per CDNA5 ISA 7.12.2:
//   A 16x4 : lanes 0-15 hold {K=0,K=1}, lanes 16-31 hold {K=2,K=3}, M = lane&15
//   B 4x16 : lanes 0-15 hold rows K=0/K=1, lanes 16-31 rows K=2/K=3, N = lane&15
//   C/D    : VGPR r -> M = r (lanes 0-15) or M = r+8 (lanes 16-31), N = lane&15
// W is staged into LDS pre-packed as float2 K-pairs (W[k][n], W[k+1][n]) so
// each B fragment is a single aligned ds_load_b64 (no register shuffling).
// Row index is clamped (not predicated) so the main loop is branch-free and
// EXEC stays all-ones (required for WMMA).

template <int K>
__global__ __launch_bounds__(128)
void k_gemm(const float* __restrict__ in, const float* __restrict__ W,
            float* __restrict__ out, int n) {
    __shared__ float2 Wp[(K / 2) * HD];     // [kpair][n]
    for (int i = threadIdx.x; i < (K / 2) * HD; i += 128) {
        int kp = i / HD, c = i % HD;
        float2 v;
        v.x = W[(2 * kp)     * HD + c];
        v.y = W[(2 * kp + 1) * HD + c];
        Wp[i] = v;
    }
    __syncthreads();

    const int lane    = threadIdx.x & 31;
    const int wave    = threadIdx.x >> 5;
    const int rowBase = blockIdx.x * 64 + wave * 16;
    const int mA      = rowBase + (lane & 15);
    const int mLoad   = (mA < n) ? mA : (n - 1);   // clamp: branch-free loads
    const int kOff    = (lane >> 4) << 1;          // 0 for lanes 0-15, 2 for 16-31
    const int ncol    = lane & 15;

    const float*  aPtr = in + (size_t)mLoad * K + kOff;
    const float2* bPtr = Wp + (kOff >> 1) * HD + ncol;

    v8f acc[4] = {v8f{}, v8f{}, v8f{}, v8f{}};

#pragma unroll
    for (int k0 = 0; k0 < K; k0 += 4) {
        const v2f a = *(const v2f*)(aPtr + k0);
#pragma unroll
        for (int t = 0; t < 4; ++t) {
            const float2 bv = bPtr[(k0 >> 1) * HD + t * 16];
            v2f b; b.x = bv.x; b.y = bv.y;
            acc[t] = __builtin_amdgcn_wmma_f32_16x16x4_f32(
                false, a, false, b, (short)0, acc[t], false, false);
        }
    }

    const int mBase = rowBase + ((lane >> 4) << 3);   // +8 for lanes 16-31
#pragma unroll
    for (int t = 0; t < 4; ++t) {
#pragma unroll
        for (int r = 0; r < 8; ++r) {
            int m = mBase + r;
            if (m < n) out[(size_t)m * HD + t * 16 + ncol] = acc[t][r];
        }
    }
}

// ---------------- edge aggregation: agg[dst] += h[src] * dis[src]*dis[dst] ----
// One wave per edge; lane covers features [2*lane, 2*lane+1]. Edge id is
// forced wave-uniform via readfirstlane so index/norm loads become scalar.

__global__ __launch_bounds__(256)
void k_aggr(const float* __restrict__ h, const float* __restrict__ dis,
            const long long* __restrict__ src, const long long* __restrict__ dst,
            float* agg, int e) {
    const int wv  = __builtin_amdgcn_readfirstlane(threadIdx.x >> 5);
    const int eid = blockIdx.x * 8 + wv;
    if (eid >= e) return;
    const int lane = threadIdx.x & 31;
    const int s = (int)src[eid];
    const int d = (int)dst[eid];
    const float w = dis[s] * dis[d];
    const float2 hv = *(const float2*)(h + (size_t)s * HD + lane * 2);
    float* p = agg + (size_t)d * HD + lane * 2;
    atomic_add_f32(p,     hv.x * w);
    atomic_add_f32(p + 1, hv.y * w);
}

// ---------------- epilogue: out = [relu](agg + h*dis^2 + b) ----------------

__global__ __launch_bounds__(256)
void k_epilogue(const float* __restrict__ agg, const float* __restrict__ h,
                const float* __restrict__ dis, const float* __restrict__ bias,
                float* __restrict__ out, int n, int doRelu) {
    int i = blockIdx.x * 256 + threadIdx.x;       // one thread per float2
    if (i >= n * 32) return;
    int node = i >> 5;
    int f2   = (i & 31) * 2;
    float w  = dis[node]; w *= w;                  // self-loop norm
    float2 hv = *(const float2*)(h    + (size_t)node * HD + f2);
    float2 av = *(const float2*)(agg  + (size_t)node * HD + f2);
    float2 bv = *(const float2*)(bias + f2);
    float r0 = av.x + hv.x * w + bv.x;
    float r1 = av.y + hv.y * w + bv.y;
    if (doRelu) { r0 = fmaxf(r0, 0.0f); r1 = fmaxf(r1, 0.0f); }
    float2 res; res.x = r0; res.y = r1;
    *(float2*)(out + (size_t)node * HD + f2) = res;
}

// ---------------- driver ----------------

static inline int cdiv(long long a, long long b) { return (int)((a + b - 1) / b); }

extern "C" void kernel_launch(void* const* d_in, const int* in_sizes, int n_in,
                              void* d_out, int out_size, void* d_ws, size_t ws_size,
                              hipStream_t stream) {
    const float*     x  = (const float*)d_in[0];
    const long long* ei = (const long long*)d_in[1];
    const float* W1 = (const float*)d_in[2];
    const float* b1 = (const float*)d_in[3];
    const float* W2 = (const float*)d_in[4];
    const float* b2 = (const float*)d_in[5];
    const float* W3 = (const float*)d_in[6];
    const float* b3 = (const float*)d_in[7];

    const int N = in_sizes[0] / 128;   // 100000
    const int E = in_sizes[1] / 2;     // 3200000
    const long long* src = ei;         // edge_index[0]
    const long long* dst = ei + E;     // edge_index[1]

    float* ws   = (float*)d_ws;
    float* deg  = ws;                               // N
    float* dis  = deg + N;                          // N
    float* hbuf = dis + N;                          // N*64
    float* agg  = hbuf + (size_t)N * HD;            // N*64
    float* act  = agg  + (size_t)N * HD;            // N*64
    float* outp = (float*)d_out;

    const size_t NH = (size_t)N * HD;
    const int zgrid = 2048;

    // symmetric normalization D^-1/2 (A+I) D^-1/2
    k_init_deg<<<cdiv(N, 256), 256, 0, stream>>>(deg, N);
    k_deg     <<<cdiv(E, 256), 256, 0, stream>>>(dst, deg, E);
    k_dis     <<<cdiv(N, 256), 256, 0, stream>>>(deg, dis, N);

    // ---- layer 1: K=128, relu ----
    k_gemm<128><<<cdiv(N, 64), 128, 0, stream>>>(x, W1, hbuf, N);
    k_zero<<<zgrid, 256, 0, stream>>>(agg, NH);
    k_aggr<<<cdiv(E, 8), 256, 0, stream>>>(hbuf, dis, src, dst, agg, E);
    k_epilogue<<<cdiv((long long)N * 32, 256), 256, 0, stream>>>(agg, hbuf, dis, b1, act, N, 1);

    // ---- layer 2: K=64, relu ----
    k_gemm<64><<<cdiv(N, 64), 128, 0, stream>>>(act, W2, hbuf, N);
    k_zero<<<zgrid, 256, 0, stream>>>(agg, NH);
    k_aggr<<<cdiv(E, 8), 256, 0, stream>>>(hbuf, dis, src, dst, agg, E);
    k_epilogue<<<cdiv((long long)N * 32, 256), 256, 0, stream>>>(agg, hbuf, dis, b2, act, N, 1);

    // ---- layer 3: K=64, no relu, write d_out ----
    k_gemm<64><<<cdiv(N, 64), 128, 0, stream>>>(act, W3, hbuf, N);
    k_zero<<<zgrid, 256, 0, stream>>>(agg, NH);
    k_aggr<<<cdiv(E, 8), 256, 0, stream>>>(hbuf, dis, src, dst, agg, E);
    k_epilogue<<<cdiv((long long)N * 32, 256), 256, 0, stream>>>(agg, hbuf, dis, b3, outp, N, 0);
}